// GNNStack_6425271075235
// MI455X (gfx1250) — compile-verified
//
#include <hip/hip_runtime.h>
#include <math.h>

#define NNODES 50000
#define NEDGES 800000
#define NPAD   50048   // 391 * 128

typedef float v2f __attribute__((ext_vector_type(2)));
typedef float v8f __attribute__((ext_vector_type(8)));

// --------------------------------------------------------------------------
// CDNA5 async global->LDS path (ASYNCcnt), with fallback to register staging.
// Builtin signature (from toolchain diagnostic): (v4i AS1*, v4i AS3*, Ii, Ii).
// --------------------------------------------------------------------------
#if defined(__has_builtin)
#if __has_builtin(__builtin_amdgcn_global_load_async_to_lds_b128) && \
    __has_builtin(__builtin_amdgcn_s_wait_asynccnt)
#define USE_ASYNC_LDS 1
#endif
#endif

#ifdef USE_ASYNC_LDS
typedef int v4i __attribute__((vector_size(16)));
__device__ __forceinline__ void async_b128(const float* g, float* l) {
  __builtin_amdgcn_global_load_async_to_lds_b128(
      (__attribute__((address_space(1))) v4i*)g,
      (__attribute__((address_space(3))) v4i*)l, 0, 0);
}
#endif

// ---------------------------------------------------------------------------
// WMMA fp32 GEMM:  C[M,BN] = act(A[M,K] @ W[K,BN] + bias),  K = 128 or 256.
// TWO_A=true: A = concat(A0, A1) along K (each K=128, row pitch 128);
// W row-major [K, BN]. Block: 256 thr = 8 waves; BM=128; wave = 16 x BN strip.
// A tile double-buffered in LDS (async loads when available); W tile held
// transposed in LDS so each B fragment is a single ds_load_2addr_b32 pair.
// ---------------------------------------------------------------------------
template <int BN, bool RELU, bool TWO_A>
__global__ __launch_bounds__(256) void gemm_wmma(
    const float* __restrict__ A0, const float* __restrict__ A1,
    const float* __restrict__ W, const float* __restrict__ bias,
    float* __restrict__ C, int M) {
  constexpr int BM = 128, KC = 32;
  constexpr int NT  = BN / 16;                  // WMMA tiles along N per wave
  constexpr int NCH = (TWO_A ? 2 : 1) * (128 / KC);
  constexpr int WLD = (KC * BN) / 1024;         // float4 W loads per thread

  __shared__ float As[2][BM][KC + 4];           // pitch 36: 16B-aligned rows
  __shared__ float Wt[BN][KC + 1];              // transposed W tile, pitch 33

  const int tid   = threadIdx.x;
  const int wave  = tid >> 5;
  const int lane  = tid & 31;
  const int lhalf = lane >> 4;                  // 0: lanes 0-15, 1: lanes 16-31
  const int l16   = lane & 15;
  const int m0    = blockIdx.x * BM;
  const int wrow  = wave * 16;

  v8f acc[NT] = {};
  float4 wreg[WLD];
#ifndef USE_ASYNC_LDS
  float4 areg[4];
#endif

  // ---- issue global traffic for chunk c (A -> LDS async / regs, W -> regs)
  auto issue = [&](int c) {
    const float* __restrict__ A = (TWO_A && c >= 4) ? A1 : A0;
    const int k0    = (c & 3) * KC;
    const int krow0 = TWO_A ? ((c >> 2) * 128) : 0;
    const int buf   = c & 1;
    (void)buf;
#pragma unroll
    for (int i = 0; i < 4; ++i) {
      int idx = tid + i * 256;                  // 0..1023 float4 slots
      int r = idx >> 3, c4 = (idx & 7) * 4;
      if (m0 + r < M) {                         // padded rows: stale LDS is OK
#ifdef USE_ASYNC_LDS
        async_b128(A + (size_t)(m0 + r) * 128 + k0 + c4, &As[buf][r][c4]);
#else
        areg[i] = *(const float4*)(A + (size_t)(m0 + r) * 128 + k0 + c4);
#endif
      }
    }
#pragma unroll
    for (int i = 0; i < WLD; ++i) {
      int idx = tid + i * 256;
      int r = idx / (BN / 4), c4 = (idx % (BN / 4)) * 4;
      wreg[i] = *(const float4*)(W + (size_t)(krow0 + k0 + r) * BN + c4);
    }
  };

  // ---- make chunk c visible in LDS and synchronize the block
  auto commit = [&](int c) {
    const int buf = c & 1;
    (void)buf;
#ifndef USE_ASYNC_LDS
#pragma unroll
    for (int i = 0; i < 4; ++i) {
      int idx = tid + i * 256;
      int r = idx >> 3, c4 = (idx & 7) * 4;
      if (m0 + r < M) {
        As[buf][r][c4 + 0] = areg[i].x; As[buf][r][c4 + 1] = areg[i].y;
        As[buf][r][c4 + 2] = areg[i].z; As[buf][r][c4 + 3] = areg[i].w;
      }
    }
#endif
#pragma unroll
    for (int i = 0; i < WLD; ++i) {
      int idx = tid + i * 256;
      int r = idx / (BN / 4), c4 = (idx % (BN / 4)) * 4;
      Wt[c4 + 0][r] = wreg[i].x; Wt[c4 + 1][r] = wreg[i].y;
      Wt[c4 + 2][r] = wreg[i].z; Wt[c4 + 3][r] = wreg[i].w;
    }
#ifdef USE_ASYNC_LDS
    __builtin_amdgcn_s_wait_asynccnt(0);
#endif
    __syncthreads();
  };

  // ---- 64 WMMAs on the resident chunk
  auto compute = [&](int c) {
    const int buf = c & 1;
    const float* ap = &As[buf][wrow + l16][2 * lhalf];
#pragma unroll
    for (int kk = 0; kk < KC; kk += 4) {
      v2f a;                                    // A frag: K {0|2},{1|3} halves
      a.x = ap[kk + 0];
      a.y = ap[kk + 1];
#pragma unroll
      for (int nt = 0; nt < NT; ++nt) {
        v2f b;                                  // B frag: one 2addr pair
        b.x = Wt[nt * 16 + l16][kk + 0 + lhalf];
        b.y = Wt[nt * 16 + l16][kk + 2 + lhalf];
        acc[nt] = __builtin_amdgcn_wmma_f32_16x16x4_f32(
            false, a, false, b, (short)0, acc[nt], false, false);
      }
    }
  };

  issue(0);
  commit(0);
  for (int c = 0; c < NCH; ++c) {
    if (c + 1 < NCH) issue(c + 1);              // overlap with compute(c)
    compute(c);
    __syncthreads();                            // done reading Wt / As[c&1]
    if (c + 1 < NCH) commit(c + 1);
  }

  // ---- epilogue: bias + activation + store (C/D layout: M = r + 8*lhalf) --
#pragma unroll
  for (int nt = 0; nt < NT; ++nt) {
    float bv = bias[nt * 16 + l16];
#pragma unroll
    for (int r = 0; r < 8; ++r) {
      int grow = m0 + wrow + 8 * lhalf + r;
      if (grow < M) {
        float v = acc[nt][r] + bv;
        if (RELU) v = fmaxf(v, 0.f);
        C[(size_t)grow * BN + nt * 16 + l16] = v;
      }
    }
  }
}

// ---------------------------------------------------------------------------
// One wave per edge: gather h[src] (float4/lane), atomic-add into agg[dst],
// lane 0 bumps the degree counter. Working set lives in the 192MB L2.
// ---------------------------------------------------------------------------
__global__ __launch_bounds__(256) void scatter_edges(
    const float* __restrict__ h, const int* __restrict__ src,
    const int* __restrict__ dst, float* __restrict__ agg,
    float* __restrict__ deg, int E) {
  int wid = (blockIdx.x * 256 + threadIdx.x) >> 5;
  int lane = threadIdx.x & 31;
  if (wid >= E) return;
  int s = src[wid];
  int d = dst[wid];
  const float4 v = *(const float4*)(h + (size_t)s * 128 + lane * 4);
  float* out = agg + (size_t)d * 128 + lane * 4;
  atomicAdd(out + 0, v.x);
  atomicAdd(out + 1, v.y);
  atomicAdd(out + 2, v.z);
  atomicAdd(out + 3, v.w);
  if (lane == 0) atomicAdd(deg + d, 1.0f);
}

__global__ __launch_bounds__(256) void mean_div(float* __restrict__ agg,
                                                const float* __restrict__ deg,
                                                int N) {
  int i = blockIdx.x * 256 + threadIdx.x;
  if (i >= N * 128) return;
  float dg = deg[i >> 7];
  agg[i] = agg[i] / fmaxf(dg, 1.0f);
}

// One wave per row: L2 normalize 128 floats (inputs are >= 0 from ReLU, so
// the reference's trailing ReLU is the identity).
__global__ __launch_bounds__(256) void l2norm_rows(float* __restrict__ x, int N) {
  int wid = (blockIdx.x * 256 + threadIdx.x) >> 5;
  int lane = threadIdx.x & 31;
  if (wid >= N) return;
  float4 v = *(float4*)(x + (size_t)wid * 128 + lane * 4);
  float ss = v.x * v.x + v.y * v.y + v.z * v.z + v.w * v.w;
#pragma unroll
  for (int off = 16; off > 0; off >>= 1) ss += __shfl_xor(ss, off, 32);
  float inv = 1.0f / fmaxf(sqrtf(ss), 1e-12f);
  v.x *= inv; v.y *= inv; v.z *= inv; v.w *= inv;
  *(float4*)(x + (size_t)wid * 128 + lane * 4) = v;
}

// One wave per row: log_softmax over 64 logits (2 per lane).
__global__ __launch_bounds__(256) void logsoftmax64(const float* __restrict__ z,
                                                    float* __restrict__ out,
                                                    int N) {
  int wid = (blockIdx.x * 256 + threadIdx.x) >> 5;
  int lane = threadIdx.x & 31;
  if (wid >= N) return;
  float2 v = *(const float2*)(z + (size_t)wid * 64 + lane * 2);
  float m = fmaxf(v.x, v.y);
#pragma unroll
  for (int off = 16; off > 0; off >>= 1) m = fmaxf(m, __shfl_xor(m, off, 32));
  float s = __expf(v.x - m) + __expf(v.y - m);
#pragma unroll
  for (int off = 16; off > 0; off >>= 1) s += __shfl_xor(s, off, 32);
  float lse = m + __logf(s);
  float2 o = make_float2(v.x - lse, v.y - lse);
  *(float2*)(out + (size_t)wid * 64 + lane * 2) = o;
}

// ---------------------------------------------------------------------------
extern "C" void kernel_launch(void* const* d_in, const int* in_sizes, int n_in,
                              void* d_out, int out_size, void* d_ws,
                              size_t ws_size, hipStream_t stream) {
  const int N = NNODES;
  const int E = in_sizes[1] / 2;

  const float* x   = (const float*)d_in[0];
  const int*   ei  = (const int*)d_in[1];
  const int*   src = ei;
  const int*   dst = ei + E;

  const float* W_lin[3] = {(const float*)d_in[2],  (const float*)d_in[6],
                           (const float*)d_in[10]};
  const float* b_lin[3] = {(const float*)d_in[3],  (const float*)d_in[7],
                           (const float*)d_in[11]};
  const float* W_agg[3] = {(const float*)d_in[4],  (const float*)d_in[8],
                           (const float*)d_in[12]};
  const float* b_agg[3] = {(const float*)d_in[5],  (const float*)d_in[9],
                           (const float*)d_in[13]};
  const float* W_mp1 = (const float*)d_in[14];
  const float* b_mp1 = (const float*)d_in[15];
  const float* W_mp2 = (const float*)d_in[16];
  const float* b_mp2 = (const float*)d_in[17];

  float* ws     = (float*)d_ws;
  float* bufH   = ws;                           // [NPAD,128]
  float* bufAgg = bufH   + (size_t)NPAD * 128;  // [NPAD,128]
  float* bufO0  = bufAgg + (size_t)NPAD * 128;  // [NPAD,128]
  float* bufO1  = bufO0  + (size_t)NPAD * 128;  // [NPAD,128]
  float* deg    = bufO1  + (size_t)NPAD * 128;  // [NPAD]

  const int gM       = NPAD / 128;              // 391 M-tiles
  const int gEdges   = (E + 7) / 8;             // 8 waves (edges) per block
  const int gElem128 = (N * 128 + 255) / 256;
  const int gRows    = (N + 7) / 8;             // wave-per-row kernels

  const float* cur = x;
  for (int l = 0; l < 3; ++l) {
    // h = relu(cur @ W_lin + b_lin)
    gemm_wmma<128, true, false><<<gM, 256, 0, stream>>>(
        cur, nullptr, W_lin[l], b_lin[l], bufH, N);
    // mean aggregation over edges
    (void)hipMemsetAsync(bufAgg, 0, (size_t)N * 128 * sizeof(float), stream);
    (void)hipMemsetAsync(deg, 0, (size_t)N * sizeof(float), stream);
    scatter_edges<<<gEdges, 256, 0, stream>>>(bufH, src, dst, bufAgg, deg, E);
    mean_div<<<gElem128, 256, 0, stream>>>(bufAgg, deg, N);
    // out = l2norm(relu(concat(cur, agg) @ W_agg + b_agg))
    float* outb = (l & 1) ? bufO1 : bufO0;
    gemm_wmma<128, true, true><<<gM, 256, 0, stream>>>(
        cur, bufAgg, W_agg[l], b_agg[l], outb, N);
    l2norm_rows<<<gRows, 256, 0, stream>>>(outb, N);
    cur = outb;
  }

  // post_mp: two GEMMs then log_softmax
  gemm_wmma<128, false, false><<<gM, 256, 0, stream>>>(
      cur, nullptr, W_mp1, b_mp1, bufH, N);
  gemm_wmma<64, false, false><<<gM, 256, 0, stream>>>(
      bufH, nullptr, W_mp2, b_mp2, bufAgg, N);
  logsoftmax64<<<gRows, 256, 0, stream>>>(bufAgg, (float*)d_out, N);
}